// GNN5_50483045597220
// MI455X (gfx1250) — compile-verified
//
#include <hip/hip_runtime.h>
#include <hip/hip_bf16.h>
#include <math.h>

typedef float v2f __attribute__((ext_vector_type(2)));
typedef float v8f __attribute__((ext_vector_type(8)));

// ---------------------------------------------------------------------------
// Utility kernels
// ---------------------------------------------------------------------------
__global__ void k_fill_f32(float* p, float v, int n) {
    int i = blockIdx.x * blockDim.x + threadIdx.x;
    if (i < n) p[i] = v;
}
__global__ void k_fill_u32(unsigned* p, unsigned v, int n) {
    int i = blockIdx.x * blockDim.x + threadIdx.x;
    if (i < n) p[i] = v;
}
__global__ void k_copy_f32(const float* __restrict__ a, float* __restrict__ b, int n) {
    int i = blockIdx.x * blockDim.x + threadIdx.x;
    if (i < n) b[i] = a[i];
}
// out[n,F] = broadcast bias row
__global__ void k_bias(float* __restrict__ out, const float* __restrict__ b, int N, int F) {
    int i = blockIdx.x * blockDim.x + threadIdx.x;
    if (i < N * F) out[i] = b[i % F];
}

// ---------------------------------------------------------------------------
// Graph preprocessing (input-only, computed once)
// ---------------------------------------------------------------------------
__global__ void k_deg(const int* __restrict__ ei, const float* __restrict__ ea,
                      float* __restrict__ deg, int E, int N) {
    int e = blockIdx.x * blockDim.x + threadIdx.x;
    if (e >= E + N) return;
    int d; float w;
    if (e < E) { d = ei[E + e]; w = ea[e]; }
    else       { d = e - E;     w = 1.0f; }
    atomicAdd(&deg[d], w);
}
__global__ void k_dinv(float* deg, int N) {
    int i = blockIdx.x * blockDim.x + threadIdx.x;
    if (i >= N) return;
    float d = deg[i];
    deg[i] = (d > 0.0f) ? (1.0f / sqrtf(d)) : 0.0f;
}
__global__ void k_norm(const int* __restrict__ ei, const float* __restrict__ ea,
                       const float* __restrict__ dinv, float* __restrict__ norm,
                       int E, int N) {
    int e = blockIdx.x * blockDim.x + threadIdx.x;
    if (e >= E + N) return;
    int s, d; float w;
    if (e < E) { s = ei[e]; d = ei[E + e]; w = ea[e]; }
    else       { s = e - E; d = e - E;     w = 1.0f; }
    norm[e] = dinv[s] * w * dinv[d];
}

// ---------------------------------------------------------------------------
// FP32 WMMA GEMM: C[n,F] = A[n,K] * B[K,F], K % 4 == 0.
// One wave per 16x16 output tile; V_WMMA_F32_16X16X4_F32 accumulation.
// Wave-uniform fast path (full tile): unguarded b64 A loads + b32 B loads,
// EXEC stays all-ones throughout (required by WMMA), no branches in the loop.
// Slow path (edge tiles): clamped addresses + 0/1 mask multiply, branch-free.
// A frag: lane<16 holds (M=lane, K=k..k+1), lane>=16 holds (M=lane-16, K=k+2..k+3)
// B frag: lane<16 holds (K=k..k+1, N=lane), lane>=16 holds (K=k+2..k+3, N=lane-16)
// C/D:   VGPR i -> row i (lanes 0-15) / row i+8 (lanes 16-31)
// ---------------------------------------------------------------------------
__global__ void k_gemm_wmma(const float* __restrict__ A, const float* __restrict__ B,
                            float* __restrict__ C, int n, int K, int F, int ct) {
    int gtid = blockIdx.x * blockDim.x + threadIdx.x;
    int wid  = gtid >> 5;
    int lane = threadIdx.x & 31;
    int tr = wid / ct, tc = wid - tr * ct;
    int row0 = tr * 16;
    if (row0 >= n) return;                 // wave-uniform: EXEC all-ones below
    int col0 = tc * 16;
    int half = lane >> 4;                  // 0 or 1
    int l16  = lane & 15;
    int ar = row0 + l16;
    int bc = col0 + l16;
    v8f acc = {};

    if (row0 + 16 <= n && col0 + 16 <= F) {        // wave-uniform fast path
        const float* ap = A + (size_t)ar * K + 2 * half;   // even offset: 8B aligned
        const float* bp = B + (size_t)(2 * half) * F + bc;
        for (int k = 0; k < K; k += 4) {
            v2f a = *(const v2f*)ap;               // K = k+2h, k+2h+1
            v2f b; b.x = bp[0]; b.y = bp[F];       // K = k+2h, k+2h+1 at col bc
            acc = __builtin_amdgcn_wmma_f32_16x16x4_f32(
                false, a, false, b, (short)0, acc, false, false);
            ap += 4;
            bp += 4 * (size_t)F;
        }
    } else {                                        // edge tiles: branch-free masks
        bool arok = (ar < n), bcok = (bc < F);
        float am = arok ? 1.0f : 0.0f;
        float bm = bcok ? 1.0f : 0.0f;
        const float* ap = A + (size_t)(arok ? ar : (n - 1)) * K + 2 * half;
        const float* bp = B + (size_t)(2 * half) * F + (bcok ? bc : 0);
        for (int k = 0; k < K; k += 4) {
            v2f a; a.x = ap[0] * am; a.y = ap[1] * am;
            v2f b; b.x = bp[0] * bm; b.y = bp[F] * bm;
            acc = __builtin_amdgcn_wmma_f32_16x16x4_f32(
                false, a, false, b, (short)0, acc, false, false);
            ap += 4;
            bp += 4 * (size_t)F;
        }
    }

    if (bc < F) {
#pragma unroll
        for (int i = 0; i < 8; ++i) {
            int r = row0 + i + 8 * half;
            if (r < n) C[r * F + bc] = acc[i];
        }
    }
}

// ---------------------------------------------------------------------------
// Attention scalars, wave-per-node (coalesced): es[n]=xs[n,:]·asrc, ed likewise
// ---------------------------------------------------------------------------
__global__ void k_es_ed(const float* __restrict__ xs, const float* __restrict__ asrc,
                        const float* __restrict__ adst, float* __restrict__ es,
                        float* __restrict__ ed, int N, int F) {
    int wid  = (blockIdx.x * blockDim.x + threadIdx.x) >> 5;
    int lane = threadIdx.x & 31;
    if (wid >= N) return;
    const float* row = xs + (size_t)wid * F;
    float s = 0.0f, d = 0.0f;
    for (int f = lane; f < F; f += 32) {
        float v = row[f];
        s += v * asrc[f];
        d += v * adst[f];
    }
    for (int o = 16; o > 0; o >>= 1) {
        s += __shfl_down(s, o, 32);
        d += __shfl_down(d, o, 32);
    }
    if (lane == 0) { es[wid] = s; ed[wid] = d; }
}

// Order-preserving float <-> u32 for integer atomicMax
__device__ __forceinline__ unsigned f32_ord(float f) {
    unsigned u = __float_as_uint(f);
    return (u & 0x80000000u) ? ~u : (u | 0x80000000u);
}
__device__ __forceinline__ float ord_f32(unsigned u) {
    return __uint_as_float((u & 0x80000000u) ? (u & 0x7FFFFFFFu) : ~u);
}

__global__ void k_alpha_max(const int* __restrict__ ei, const float* __restrict__ es,
                            const float* __restrict__ ed, float* __restrict__ alpha,
                            unsigned* __restrict__ m, int E, int N) {
    int e = blockIdx.x * blockDim.x + threadIdx.x;
    if (e >= E + N) return;
    int s, d;
    if (e < E) { s = ei[e]; d = ei[E + e]; } else { s = e - E; d = s; }
    float v = es[s] + ed[d];
    v = (v > 0.0f) ? v : 0.2f * v;         // leaky_relu(0.2)
    alpha[e] = v;
    atomicMax(&m[d], f32_ord(v));
}
__global__ void k_exp_sum(const int* __restrict__ ei, float* __restrict__ alpha,
                          const unsigned* __restrict__ m, float* __restrict__ ssum,
                          int E, int N) {
    int e = blockIdx.x * blockDim.x + threadIdx.x;
    if (e >= E + N) return;
    int d = (e < E) ? ei[E + e] : (e - E);
    float a = expf(alpha[e] - ord_f32(m[d]));
    alpha[e] = a;
    atomicAdd(&ssum[d], a);
}
__global__ void k_coef(const int* __restrict__ ei, float* __restrict__ alpha,
                       const float* __restrict__ ssum, int E, int N) {
    int e = blockIdx.x * blockDim.x + threadIdx.x;
    if (e >= E + N) return;
    int d = (e < E) ? ei[E + e] : (e - E);
    alpha[e] = alpha[e] / ssum[d];
}
// out[dst,f] += xs[src,f] * coef[e]  (thread per (edge, feature): coalesced)
__global__ void k_aggr(const int* __restrict__ ei, const float* __restrict__ coef,
                       const float* __restrict__ xs, float* __restrict__ out,
                       int E, int N, int F) {
    int tid = blockIdx.x * blockDim.x + threadIdx.x;
    if (tid >= (E + N) * F) return;
    int e = tid / F, f = tid - e * F;
    int s, d;
    if (e < E) { s = ei[e]; d = ei[E + e]; } else { s = e - E; d = s; }
    atomicAdd(&out[(size_t)d * F + f], xs[(size_t)s * F + f] * coef[e]);
}

// temp[n,:] = [x(TP) | h0(H) | h1(H)]
__global__ void k_concat(const float* __restrict__ x, const float* __restrict__ h0,
                         const float* __restrict__ h1, float* __restrict__ tmp,
                         int N, int TP, int H, int T) {
    int tid = blockIdx.x * blockDim.x + threadIdx.x;
    if (tid >= N * T) return;
    int n = tid / T, f = tid - n * T;
    float v;
    if (f < TP)          v = x[n * TP + f];
    else if (f < TP + H) v = h0[n * H + (f - TP)];
    else                 v = h1[n * H + (f - TP - H)];
    tmp[tid] = v;
}

// Head GEMV, wave-per-node (coalesced): xs2[n] = xg[n,:]·W2 ; es/ed scalar-scaled
__global__ void k_gemv(const float* __restrict__ xg, const float* __restrict__ w,
                       const float* __restrict__ asrc, const float* __restrict__ adst,
                       float* __restrict__ xs2, float* __restrict__ es,
                       float* __restrict__ ed, int N, int K) {
    int wid  = (blockIdx.x * blockDim.x + threadIdx.x) >> 5;
    int lane = threadIdx.x & 31;
    if (wid >= N) return;
    const float* row = xg + (size_t)wid * K;
    float acc = 0.0f;
    for (int k = lane; k < K; k += 32) acc += row[k] * w[k];
    for (int o = 16; o > 0; o >>= 1) acc += __shfl_down(acc, o, 32);
    if (lane == 0) {
        xs2[wid] = acc;
        es[wid]  = acc * asrc[0];
        ed[wid]  = acc * adst[0];
    }
}

// Clip, emit prediction column, autoregressive shift
__global__ void k_finalize(const float* __restrict__ out2, const float* __restrict__ xcur,
                           float* __restrict__ xnext, float* __restrict__ y,
                           int N, int TP, int TF, int t) {
    int n = blockIdx.x * blockDim.x + threadIdx.x;
    if (n >= N) return;
    float v = out2[n];
    v = fminf(fmaxf(v, 0.0f), 90.0f);
    y[n * TF + t] = v;
    for (int j = 0; j < TP - 1; ++j) xnext[n * TP + j] = xcur[n * TP + j + 1];
    xnext[n * TP + TP - 1] = v;
}

// ---------------------------------------------------------------------------
static inline int cdiv(int a, int b) { return (a + b - 1) / b; }

extern "C" void kernel_launch(void* const* d_in, const int* in_sizes, int n_in,
                              void* d_out, int out_size, void* d_ws, size_t ws_size,
                              hipStream_t stream) {
    const float* x   = (const float*)d_in[0];
    const int*   ei  = (const int*)  d_in[2];
    const float* ea  = (const float*)d_in[3];
    const float* W0  = (const float*)d_in[4];
    const float* as0 = (const float*)d_in[5];
    const float* ad0 = (const float*)d_in[6];
    const float* b0  = (const float*)d_in[7];
    const float* W1  = (const float*)d_in[8];
    const float* as1 = (const float*)d_in[9];
    const float* ad1 = (const float*)d_in[10];
    const float* b1  = (const float*)d_in[11];
    const float* Wg  = (const float*)d_in[12];
    const float* bg  = (const float*)d_in[13];
    const float* W2  = (const float*)d_in[14];
    const float* as2 = (const float*)d_in[15];
    const float* ad2 = (const float*)d_in[16];
    const float* b2  = (const float*)d_in[17];
    float* y = (float*)d_out;

    const int TP   = 12;
    const int N    = in_sizes[0] / TP;
    const int E    = in_sizes[2] / 2;
    const int HID  = in_sizes[7];    // 64
    const int TEMP = in_sizes[13];   // 140
    const int TF   = in_sizes[1] / N;
    const int EL   = E + N;

    // bump allocator over d_ws (all block sizes even -> 8B alignment preserved)
    float* wsp = (float*)d_ws;
    size_t off = 0;
    auto alloc = [&](size_t cnt) { float* p = wsp + off; off += (cnt + 1) & ~(size_t)1; return p; };
    float*    xA    = alloc((size_t)N * TP);
    float*    xB    = alloc((size_t)N * TP);
    float*    dinv  = alloc(N);          // holds degree, then 1/sqrt(degree)
    float*    norm  = alloc(EL);
    float*    alpha = alloc(EL);
    unsigned* mmax  = (unsigned*)alloc(N);
    float*    ssum  = alloc(N);
    float*    es    = alloc(N);
    float*    ed    = alloc(N);
    float*    xs    = alloc((size_t)N * HID);
    float*    h0    = alloc((size_t)N * HID);
    float*    h1    = alloc((size_t)N * HID);
    float*    temp  = alloc((size_t)N * TEMP);
    float*    xw    = alloc((size_t)N * TEMP);
    float*    xg    = temp;              // temp dead after GEMM into xw -> reuse
    float*    xs2   = alloc(N);
    float*    out2  = alloc(N);

    const int TB = 256;
    auto gemm = [&](const float* A, const float* B, float* C, int n, int K, int F) {
        int rt = cdiv(n, 16), ct = cdiv(F, 16);
        int blocks = cdiv(rt * ct * 32, TB);
        k_gemm_wmma<<<blocks, TB, 0, stream>>>(A, B, C, n, K, F, ct);
    };
    auto gat_softmax = [&](const float* es_, const float* ed_) {
        k_fill_u32<<<cdiv(N, TB), TB, 0, stream>>>(mmax, 0u, N);
        k_fill_f32<<<cdiv(N, TB), TB, 0, stream>>>(ssum, 0.0f, N);
        k_alpha_max<<<cdiv(EL, TB), TB, 0, stream>>>(ei, es_, ed_, alpha, mmax, E, N);
        k_exp_sum<<<cdiv(EL, TB), TB, 0, stream>>>(ei, alpha, mmax, ssum, E, N);
        k_coef<<<cdiv(EL, TB), TB, 0, stream>>>(ei, alpha, ssum, E, N);
    };

    // ---- graph-invariant precompute -------------------------------------
    k_fill_f32<<<cdiv(N, TB), TB, 0, stream>>>(dinv, 0.0f, N);
    k_deg<<<cdiv(EL, TB), TB, 0, stream>>>(ei, ea, dinv, E, N);
    k_dinv<<<cdiv(N, TB), TB, 0, stream>>>(dinv, N);
    k_norm<<<cdiv(EL, TB), TB, 0, stream>>>(ei, ea, dinv, norm, E, N);
    k_copy_f32<<<cdiv(N * TP, TB), TB, 0, stream>>>(x, xA, N * TP);

    // ---- autoregressive time loop ---------------------------------------
    for (int t = 0; t < TF; ++t) {
        float* xc = (t & 1) ? xB : xA;
        float* xn = (t & 1) ? xA : xB;

        // GAT0: [N,TP] -> [N,HID]
        gemm(xc, W0, xs, N, TP, HID);
        k_es_ed<<<cdiv(N * 32, TB), TB, 0, stream>>>(xs, as0, ad0, es, ed, N, HID);
        gat_softmax(es, ed);
        k_bias<<<cdiv(N * HID, TB), TB, 0, stream>>>(h0, b0, N, HID);
        k_aggr<<<cdiv(EL * HID, TB), TB, 0, stream>>>(ei, alpha, xs, h0, E, N, HID);

        // GAT1: [N,HID] -> [N,HID]
        gemm(h0, W1, xs, N, HID, HID);
        k_es_ed<<<cdiv(N * 32, TB), TB, 0, stream>>>(xs, as1, ad1, es, ed, N, HID);
        gat_softmax(es, ed);
        k_bias<<<cdiv(N * HID, TB), TB, 0, stream>>>(h1, b1, N, HID);
        k_aggr<<<cdiv(EL * HID, TB), TB, 0, stream>>>(ei, alpha, xs, h1, E, N, HID);

        // GCN over concat(x, h0, h1): [N,TEMP] -> [N,TEMP]
        k_concat<<<cdiv(N * TEMP, TB), TB, 0, stream>>>(xc, h0, h1, temp, N, TP, HID, TEMP);
        gemm(temp, Wg, xw, N, TEMP, TEMP);
        k_bias<<<cdiv(N * TEMP, TB), TB, 0, stream>>>(xg, bg, N, TEMP);
        k_aggr<<<cdiv(EL * TEMP, TB), TB, 0, stream>>>(ei, norm, xw, xg, E, N, TEMP);

        // GAT head: [N,TEMP] -> [N,1], clip, shift
        k_gemv<<<cdiv(N * 32, TB), TB, 0, stream>>>(xg, W2, as2, ad2, xs2, es, ed, N, TEMP);
        gat_softmax(es, ed);
        k_bias<<<cdiv(N, TB), TB, 0, stream>>>(out2, b2, N, 1);
        k_aggr<<<cdiv(EL, TB), TB, 0, stream>>>(ei, alpha, xs2, out2, E, N, 1);
        k_finalize<<<cdiv(N, TB), TB, 0, stream>>>(out2, xc, xn, y, N, TP, TF, t);
    }
}